// GNNLayer_22119081574562
// MI455X (gfx1250) — compile-verified
//
#include <hip/hip_runtime.h>
#include <hip/hip_bf16.h>

#define N_NODES 100000
#define IN_CH   128
#define OUT_CH  128
#define N_EDGES 800000

typedef __attribute__((ext_vector_type(2))) float v2f;
typedef __attribute__((ext_vector_type(8))) float v8f;

// ---------------------------------------------------------------------------
// Kernel 1: re-initialize workspace every launch (harness poisons ws once and
// never re-poisons; we must be deterministic). deg starts at 1.0 (self-loop),
// agg starts at 0.
// ---------------------------------------------------------------------------
__global__ void k_init(float* __restrict__ deg, float* __restrict__ agg) {
    int i = blockIdx.x * blockDim.x + threadIdx.x;
    if (i < N_NODES) deg[i] = 1.0f;
    int total = N_NODES * IN_CH;
    if (i < total) agg[i] = 0.0f;
    // note: grid sized for N_NODES*IN_CH (> N_NODES), both branches covered
}

// ---------------------------------------------------------------------------
// Kernel 2: degree of destination nodes over the 800K real edges.
// ---------------------------------------------------------------------------
__global__ void k_degree(const long long* __restrict__ dst,
                         float* __restrict__ deg) {
    int e = blockIdx.x * blockDim.x + threadIdx.x;
    if (e < N_EDGES) {
        atomicAdd(&deg[(int)dst[e]], 1.0f);
    }
}

// ---------------------------------------------------------------------------
// Kernel 3: deg -> deg^{-1/2} in place (deg >= 1 always due to self-loop).
// ---------------------------------------------------------------------------
__global__ void k_rsqrt(float* __restrict__ deg) {
    int i = blockIdx.x * blockDim.x + threadIdx.x;
    if (i < N_NODES) {
        deg[i] = rsqrtf(deg[i]);
    }
}

// ---------------------------------------------------------------------------
// Kernel 4: message scatter. One wave32 per (edge | self-loop). Lane l owns
// channels [4l, 4l+4): coalesced float4 gather of x[src], scaled by
// dinv[src]*dinv[dst], 4x global_atomic_add_f32 into agg[dst].
// L2 (192MB) holds agg (51MB) + hot x, so atomics resolve near-chip.
// ---------------------------------------------------------------------------
__global__ void k_scatter(const float* __restrict__ x,
                          const long long* __restrict__ src,
                          const long long* __restrict__ dst,
                          const float* __restrict__ dinv,
                          float* __restrict__ agg) {
    int gid  = blockIdx.x * blockDim.x + threadIdx.x;
    int wave = gid >> 5;
    int lane = gid & 31;
    if (wave >= N_EDGES + N_NODES) return;

    int s, d;
    float norm;
    if (wave < N_EDGES) {
        s = (int)src[wave];
        d = (int)dst[wave];
        norm = dinv[s] * dinv[d];
    } else {                       // self loop: norm = dinv[i]^2
        s = d = wave - N_EDGES;
        float di = dinv[s];
        norm = di * di;
    }

    const float4* xs = (const float4*)(x + (long long)s * IN_CH);
    float4 v = xs[lane];           // wave covers 512B contiguous
    float* ap = agg + (long long)d * IN_CH + lane * 4;
    atomicAdd(ap + 0, v.x * norm);
    atomicAdd(ap + 1, v.y * norm);
    atomicAdd(ap + 2, v.z * norm);
    atomicAdd(ap + 3, v.w * norm);
}

// ---------------------------------------------------------------------------
// Kernel 5: h = agg @ W + bias via V_WMMA_F32_16X16X4_F32 (full fp32 WMMA).
// Block = 256 thr = 8 waves. Block owns 16 output rows; wave w owns the
// 16x16 tile at columns [16w, 16w+16). The 16x128 agg stripe is staged in
// LDS (8KB, shared by all 8 waves). W (64KB) is read from global (L2-hot).
//
// Fragment layouts per CDNA5 ISA 7.12.2 (32-bit, wave32):
//   A (16x4):  lanes 0-15 -> M=lane,    K = v     (v = VGPR index 0,1)
//              lanes16-31 -> M=lane-16, K = v + 2
//   B (4x16):  lanes 0-15 -> N=lane,    K = v ; lanes16-31 -> N=lane-16, K=v+2
//   C/D 16x16: VGPR v: lanes 0-15 -> (M=v,   N=lane)
//                      lanes16-31 -> (M=v+8, N=lane-16)
// EXEC is all-ones in this kernel (exact tiling), as WMMA requires.
// ---------------------------------------------------------------------------
__global__ __launch_bounds__(256) void k_gemm(const float* __restrict__ agg,
                                              const float* __restrict__ W,
                                              const float* __restrict__ bias,
                                              float* __restrict__ out) {
    __shared__ float sA[16 * IN_CH];          // 8 KB stripe of agg

    const int tid  = threadIdx.x;
    const int row0 = blockIdx.x * 16;

    // cooperative stage of the 16x128 A stripe (float4, fully coalesced)
    {
        const float4* A4 = (const float4*)(agg + (long long)row0 * IN_CH);
        float4* sA4 = (float4*)sA;
        for (int i = tid; i < 16 * IN_CH / 4; i += 256) sA4[i] = A4[i];
    }
    __syncthreads();

    const int wave = tid >> 5;
    const int lane = tid & 31;
    const int col0 = wave * 16;
    const int lrow = lane & 15;               // 0..15
    const int hi   = lane >> 4;               // 0|1 -> K offset +2

    v8f acc = {};

#pragma unroll 4
    for (int k = 0; k < IN_CH; k += 4) {
        const int kb = k + hi * 2;
        // A fragment: 2 consecutive K values for row lrow
        v2f av = *(const v2f*)(sA + lrow * IN_CH + kb);
        // B fragment: W[kb+0][col0+lrow], W[kb+1][col0+lrow]
        v2f bv;
        bv.x = W[(kb + 0) * OUT_CH + col0 + lrow];
        bv.y = W[(kb + 1) * OUT_CH + col0 + lrow];
        acc = __builtin_amdgcn_wmma_f32_16x16x4_f32(
            /*neg_a=*/false, av, /*neg_b=*/false, bv,
            /*c_mod=*/(short)0, acc, /*reuse_a=*/false, /*reuse_b=*/false);
    }

    const float b  = bias[col0 + lrow];
    const int orow = row0 + hi * 8;
    const int ocol = col0 + lrow;
#pragma unroll
    for (int v = 0; v < 8; ++v) {
        out[(long long)(orow + v) * OUT_CH + ocol] = acc[v] + b;
    }
}

// ---------------------------------------------------------------------------
// Host-side launcher. Inputs (setup_inputs order):
//   d_in[0] = x          [N_NODES * IN_CH]  f32
//   d_in[1] = edge_index [2 * N_EDGES]      int64
//   d_in[2] = weight     [IN_CH * OUT_CH]   f32
//   d_in[3] = bias       [OUT_CH]           f32
// d_out = [N_NODES * OUT_CH] f32
// d_ws  = deg[N_NODES] f32  followed by  agg[N_NODES * IN_CH] f32
// ---------------------------------------------------------------------------
extern "C" void kernel_launch(void* const* d_in, const int* in_sizes, int n_in,
                              void* d_out, int out_size, void* d_ws, size_t ws_size,
                              hipStream_t stream) {
    const float*     x    = (const float*)d_in[0];
    const long long* ei   = (const long long*)d_in[1];   // int64 in reference
    const float*     W    = (const float*)d_in[2];
    const float*     bias = (const float*)d_in[3];
    float*           out  = (float*)d_out;

    const long long* src = ei;
    const long long* dst = ei + N_EDGES;

    float* deg = (float*)d_ws;                 // N_NODES floats
    float* agg = deg + N_NODES;                // N_NODES*IN_CH floats (16B-aligned)

    // 1. init workspace (deg=1.0 for self loop, agg=0)
    {
        int total = N_NODES * IN_CH;
        int blocks = (total + 255) / 256;
        k_init<<<blocks, 256, 0, stream>>>(deg, agg);
    }
    // 2. degree
    k_degree<<<(N_EDGES + 255) / 256, 256, 0, stream>>>(dst, deg);
    // 3. D^{-1/2}
    k_rsqrt<<<(N_NODES + 255) / 256, 256, 0, stream>>>(deg);
    // 4. normalized gather/scatter (one wave per edge or self-loop)
    {
        long long waves = (long long)N_EDGES + N_NODES;        // 900000
        long long threads = waves * 32;
        int blocks = (int)((threads + 255) / 256);             // 112500
        k_scatter<<<blocks, 256, 0, stream>>>(x, src, dst, deg, agg);
    }
    // 5. dense transform with f32 WMMA
    k_gemm<<<N_NODES / 16, 256, 0, stream>>>(agg, W, bias, out);
}